// GeoVAEOptimizer_86139864088835
// MI455X (gfx1250) — compile-verified
//
#include <hip/hip_runtime.h>
#include <hip/hip_bf16.h>
#include <stdint.h>

#define B_N   1024
#define D_DIM 64
#define H_DIM 128
#define E_NUM 8192
#define NMST  (B_N - 1)                    // 1023
#define NPAIR ((B_N * (B_N - 1)) / 2)      // 523776
#define NSORT (1u << 19)                   // 524288
#define SEG   1024u                        // LDS sort segment (8KB of u64)

typedef float v2f __attribute__((ext_vector_type(2)));
typedef float v8f __attribute__((ext_vector_type(8)));

// ---------------- degree / symmetric normalization ----------------
__global__ void k_init_deg(float* deg) {
  int i = blockIdx.x * blockDim.x + threadIdx.x;
  if (i < B_N) deg[i] = 1.0f;                       // self-loop contribution
}
__global__ void k_accum_deg(const int* __restrict__ dst, float* deg) {
  int e = blockIdx.x * blockDim.x + threadIdx.x;
  if (e < E_NUM) atomicAdd(&deg[dst[e]], 1.0f);
}
__global__ void k_dinv(const float* __restrict__ deg, float* dinv) {
  int i = blockIdx.x * blockDim.x + threadIdx.x;
  if (i < B_N) dinv[i] = rsqrtf(fmaxf(deg[i], 1e-12f));
}

// ---------------- f32 WMMA GEMM: OUT[M,N] = X[M,K] @ W[K,N] ----------------
// grid = (M/16, N/16), block = 32 (one wave per 16x16 tile). ldx == K, ldw == N.
__global__ void k_gemm_wmma_f32(const float* __restrict__ X,
                                const float* __restrict__ W,
                                float* __restrict__ OUT, int K, int N) {
  int lane = threadIdx.x;
  int m16  = lane & 15;
  int half = lane >> 4;
  int row0 = blockIdx.x * 16;
  int col0 = blockIdx.y * 16;
  const float* xrow = X + (size_t)(row0 + m16) * K;
  const float* wcol = W + col0 + m16;
  v8f c = {0.f, 0.f, 0.f, 0.f, 0.f, 0.f, 0.f, 0.f};
  for (int k0 = 0; k0 < K; k0 += 4) {
    int ka = k0 + 2 * half;
    v2f a, b;
    a.x = xrow[ka];
    a.y = xrow[ka + 1];
    b.x = wcol[(size_t)ka * N];
    b.y = wcol[(size_t)(ka + 1) * N];
    c = __builtin_amdgcn_wmma_f32_16x16x4_f32(false, a, false, b,
                                              (short)0, c, false, false);
  }
  float* orow = OUT + col0 + m16;
#pragma unroll
  for (int r = 0; r < 8; r++)
    orow[(size_t)(row0 + r + 8 * half) * N] = c[r];
}

// ---------------- GCN aggregation ----------------
__global__ void k_gcn_self(const float* __restrict__ xw,
                           const float* __restrict__ dinv, float* agg) {
  int t = blockIdx.x * blockDim.x + threadIdx.x;
  if (t < B_N * H_DIM) {
    int i = t >> 7;
    float di = dinv[i];
    agg[t] = xw[t] * di * di;
  }
}
__global__ void k_gcn_edge(const float* __restrict__ xw,
                           const int* __restrict__ src, const int* __restrict__ dst,
                           const float* __restrict__ dinv, float* agg) {
  int t = blockIdx.x * blockDim.x + threadIdx.x;
  if (t < E_NUM * H_DIM) {
    int e = t >> 7;
    int f = t & (H_DIM - 1);
    int s = src[e], d = dst[e];
    atomicAdd(&agg[d * H_DIM + f], xw[s * H_DIM + f] * dinv[s] * dinv[d]);
  }
}
__global__ void k_bias_relu(const float* __restrict__ in, const float* __restrict__ b,
                            float* out, int N, int total) {
  int t = blockIdx.x * blockDim.x + threadIdx.x;
  if (t < total) out[t] = fmaxf(in[t] + b[t % N], 0.f);
}

// ---------------- pairwise scores -> packed sort keys ----------------
__device__ __forceinline__ uint32_t fkey(float f) {
  uint32_t u = __float_as_uint(f);
  return (u & 0x80000000u) ? ~u : (u | 0x80000000u);   // unsigned asc == float asc
}
// block = 256 threads; grid = (B, B/256). Row A[i,:], bs1, Ws2 staged in LDS.
__global__ void k_pair_scores(const float* __restrict__ Amat,
                              const float* __restrict__ Cmat,
                              const float* __restrict__ bs1,
                              const float* __restrict__ Ws2,
                              const float* __restrict__ bs2,
                              uint64_t* __restrict__ keys) {
  __shared__ float sA[H_DIM], sB[H_DIM], sW[H_DIM];
  int i   = blockIdx.x;
  int tid = threadIdx.x;
  if (tid < H_DIM) {
    sA[tid] = Amat[i * H_DIM + tid];
    sB[tid] = bs1[tid];
    sW[tid] = Ws2[tid];
  }
  __syncthreads();
  int j = blockIdx.y * blockDim.x + tid;
  if (j > i && j < B_N) {
    const float* cj = Cmat + j * H_DIM;
    float s = bs2[0];
#pragma unroll 8
    for (int k = 0; k < H_DIM; k++)
      s += fmaxf(sA[k] + cj[k] + sB[k], 0.f) * sW[k];
    unsigned idx = (unsigned)i * (2u * B_N - i - 1u) / 2u + (unsigned)(j - i - 1);
    uint32_t pk = ((uint32_t)i << 10) | (uint32_t)j;       // monotone in linear idx
    keys[idx] = ((uint64_t)fkey(s) << 32) | (uint32_t)(~pk);
  }
}
__global__ void k_pad_keys(uint64_t* keys) {
  unsigned t = blockIdx.x * blockDim.x + threadIdx.x;
  unsigned i = NPAIR + t;
  if (i < NSORT) keys[i] = 0ull;                           // sorts last (descending)
}

// ---------------- bitonic sort (descending on u64 key) ----------------
// LDS full sort of each 1024-key segment: stages k = 2..1024 (55 passes fused).
__global__ void k_bitonic_local_sort(uint64_t* __restrict__ keys) {
  __shared__ uint64_t sh[SEG];
  unsigned t    = threadIdx.x;                             // 0..511
  unsigned base = blockIdx.x * SEG;
  sh[t]        = keys[base + t];
  sh[t + 512]  = keys[base + t + 512];
  __syncthreads();
  for (unsigned k = 2; k <= SEG; k <<= 1) {
    for (unsigned j = k >> 1; j > 0; j >>= 1) {
      unsigned i   = ((t & ~(j - 1)) << 1) | (t & (j - 1));
      unsigned ixj = i + j;
      bool desc = (((base + i) & k) == 0);
      uint64_t a = sh[i], b = sh[ixj];
      if (desc ? (a < b) : (a > b)) { sh[i] = b; sh[ixj] = a; }
      __syncthreads();
    }
  }
  keys[base + t]       = sh[t];
  keys[base + t + 512] = sh[t + 512];
}
// LDS merge tail of stage k (>= 2048): passes j = 512..1 fused.
// Direction is uniform per segment: bit log2(k) of the index is (base & k).
__global__ void k_bitonic_local_merge(uint64_t* __restrict__ keys, unsigned k) {
  __shared__ uint64_t sh[SEG];
  unsigned t    = threadIdx.x;
  unsigned base = blockIdx.x * SEG;
  bool desc = ((base & k) == 0);
  sh[t]       = keys[base + t];
  sh[t + 512] = keys[base + t + 512];
  __syncthreads();
  for (unsigned j = 512; j > 0; j >>= 1) {
    unsigned i   = ((t & ~(j - 1)) << 1) | (t & (j - 1));
    unsigned ixj = i + j;
    uint64_t a = sh[i], b = sh[ixj];
    if (desc ? (a < b) : (a > b)) { sh[i] = b; sh[ixj] = a; }
    __syncthreads();
  }
  keys[base + t]       = sh[t];
  keys[base + t + 512] = sh[t + 512];
}
// Global pass for j >= 1024.
__global__ void k_bitonic_global(uint64_t* keys, unsigned j, unsigned k) {
  unsigned i = blockIdx.x * blockDim.x + threadIdx.x;
  unsigned ixj = i ^ j;
  if (ixj > i) {
    uint64_t a = keys[i], b = keys[ixj];
    bool desc = ((i & k) == 0);
    if (desc ? (a < b) : (a > b)) { keys[i] = b; keys[ixj] = a; }
  }
}

// ---------------- single-thread Kruskal (path-halving union-find) ----------------
__global__ void k_kruskal(const uint64_t* __restrict__ keys,
                          int* parent, int* rnk, int* mi, int* mj) {
  for (int v = 0; v < B_N; v++) { parent[v] = v; rnk[v] = 0; }
  int cnt = 0;
  for (unsigned t = 0; t < NSORT && cnt < NMST; t++) {
    uint32_t packed = ~((uint32_t)keys[t]);
    int i = (packed >> 10) & 1023;
    int j = packed & 1023;
    if (i >= j) continue;                              // also skips padding
    int ri = i; while (parent[ri] != ri) { parent[ri] = parent[parent[ri]]; ri = parent[ri]; }
    int rj = j; while (parent[rj] != rj) { parent[rj] = parent[parent[rj]]; rj = parent[rj]; }
    if (ri == rj) continue;
    if (rnk[ri] < rnk[rj])      parent[ri] = rj;
    else if (rnk[ri] > rnk[rj]) parent[rj] = ri;
    else                        { parent[rj] = ri; rnk[ri]++; }
    mi[cnt] = i; mj[cnt] = j; cnt++;
  }
}

// ---------------- outputs ----------------
__global__ void k_write_edges(const int* __restrict__ mi, const int* __restrict__ mj,
                              float* out) {
  int t = blockIdx.x * blockDim.x + threadIdx.x;
  if (t < NMST) {
    out[2 * t]                = (float)mi[t];
    out[2 * t + 1]            = (float)mj[t];
    out[2 * NMST + 2 * t]     = (float)mj[t];
    out[2 * NMST + 2 * t + 1] = (float)mi[t];
  }
}
__global__ void k_gather_hp(const float* __restrict__ h2,
                            const int* __restrict__ mi, const int* __restrict__ mj,
                            float* hp) {
  int t = blockIdx.x * blockDim.x + threadIdx.x;       // over 1024*256 (pad row 1023)
  int e = t >> 8;
  int c = t & 255;
  float v = 0.f;
  if (e < NMST) v = (c < H_DIM) ? h2[mi[e] * H_DIM + c] : h2[mj[e] * H_DIM + (c - H_DIM)];
  hp[t] = v;
}
__global__ void k_bias_tanh_store(const float* __restrict__ t2,
                                  const float* __restrict__ bg2, float* out) {
  int t = blockIdx.x * blockDim.x + threadIdx.x;       // over NMST*D
  if (t < NMST * D_DIM) out[t] = tanhf(t2[t] + bg2[t % D_DIM]);
}

extern "C" void kernel_launch(void* const* d_in, const int* in_sizes, int n_in,
                              void* d_out, int out_size, void* d_ws, size_t ws_size,
                              hipStream_t stream) {
  const float* mu  = (const float*)d_in[0];
  const int*   ei  = (const int*)d_in[1];
  const int* src = ei;
  const int* dst = ei + E_NUM;
  const float* W1  = (const float*)d_in[2];
  const float* b1  = (const float*)d_in[3];
  const float* W2  = (const float*)d_in[4];
  const float* b2  = (const float*)d_in[5];
  const float* Ws1 = (const float*)d_in[6];
  const float* bs1 = (const float*)d_in[7];
  const float* Ws2 = (const float*)d_in[8];
  const float* bs2 = (const float*)d_in[9];
  const float* Wg1 = (const float*)d_in[10];
  const float* bg1 = (const float*)d_in[11];
  const float* Wg2 = (const float*)d_in[12];
  const float* bg2 = (const float*)d_in[13];

  float* out_edges = (float*)d_out;                    // [2, 2*(B-1)] as floats
  float* out_gamma = (float*)d_out + 2 * 2 * NMST;     // [B-1, D]

  // ---- workspace carve ----
  char* p = (char*)d_ws;
  float*    deg    = (float*)p;    p += 4096;
  float*    dinv   = (float*)p;    p += 4096;
  float*    xw     = (float*)p;    p += (size_t)B_N * H_DIM * 4;
  float*    agg    = (float*)p;    p += (size_t)B_N * H_DIM * 4;
  float*    h1     = (float*)p;    p += (size_t)B_N * H_DIM * 4;
  float*    h2     = (float*)p;    p += (size_t)B_N * H_DIM * 4;
  float*    Amat   = (float*)p;    p += (size_t)B_N * H_DIM * 4;
  float*    Cmat   = (float*)p;    p += (size_t)B_N * H_DIM * 4;
  uint64_t* keys   = (uint64_t*)p; p += (size_t)NSORT * 8;
  int*      mi     = (int*)p;      p += 4096;
  int*      mj     = (int*)p;      p += 4096;
  int*      parent = (int*)p;      p += 4096;
  int*      rnk    = (int*)p;      p += 4096;
  float*    hp     = (float*)p;    p += (size_t)B_N * 2 * H_DIM * 4;
  float*    t1     = (float*)p;    p += (size_t)B_N * H_DIM * 4;
  float*    t2     = (float*)p;    p += (size_t)B_N * D_DIM * 4;
  (void)ws_size; (void)n_in; (void)in_sizes; (void)out_size;

  // ---- degree / D^-1/2 ----
  k_init_deg <<<4, 256, 0, stream>>>(deg);
  k_accum_deg<<<E_NUM / 256, 256, 0, stream>>>(dst, deg);
  k_dinv     <<<4, 256, 0, stream>>>(deg, dinv);

  // ---- GCN layer 1: h1 = relu(agg(mu@W1) + b1) ----
  k_gemm_wmma_f32<<<dim3(B_N / 16, H_DIM / 16), 32, 0, stream>>>(mu, W1, xw, D_DIM, H_DIM);
  k_gcn_self<<<(B_N * H_DIM) / 256, 256, 0, stream>>>(xw, dinv, agg);
  k_gcn_edge<<<(E_NUM * H_DIM) / 256, 256, 0, stream>>>(xw, src, dst, dinv, agg);
  k_bias_relu<<<(B_N * H_DIM) / 256, 256, 0, stream>>>(agg, b1, h1, H_DIM, B_N * H_DIM);

  // ---- GCN layer 2: h2 = relu(agg(h1@W2) + b2) ----
  k_gemm_wmma_f32<<<dim3(B_N / 16, H_DIM / 16), 32, 0, stream>>>(h1, W2, xw, H_DIM, H_DIM);
  k_gcn_self<<<(B_N * H_DIM) / 256, 256, 0, stream>>>(xw, dinv, agg);
  k_gcn_edge<<<(E_NUM * H_DIM) / 256, 256, 0, stream>>>(xw, src, dst, dinv, agg);
  k_bias_relu<<<(B_N * H_DIM) / 256, 256, 0, stream>>>(agg, b2, h2, H_DIM, B_N * H_DIM);

  // ---- A = h2 @ Ws1[:H], C = h2 @ Ws1[H:] ----
  k_gemm_wmma_f32<<<dim3(B_N / 16, H_DIM / 16), 32, 0, stream>>>(h2, Ws1, Amat, H_DIM, H_DIM);
  k_gemm_wmma_f32<<<dim3(B_N / 16, H_DIM / 16), 32, 0, stream>>>(h2, Ws1 + H_DIM * H_DIM, Cmat,
                                                                 H_DIM, H_DIM);

  // ---- pairwise scores -> sort keys ----
  k_pair_scores<<<dim3(B_N, B_N / 256), 256, 0, stream>>>(Amat, Cmat, bs1, Ws2, bs2, keys);
  k_pad_keys<<<(NSORT - NPAIR + 255) / 256, 256, 0, stream>>>(keys);

  // ---- descending bitonic sort: LDS-fused stages + sparse global passes ----
  k_bitonic_local_sort<<<NSORT / SEG, 512, 0, stream>>>(keys);      // k = 2..1024
  for (unsigned k = 2 * SEG; k <= NSORT; k <<= 1) {
    for (unsigned j = k >> 1; j >= SEG; j >>= 1)
      k_bitonic_global<<<NSORT / 256, 256, 0, stream>>>(keys, j, k);
    k_bitonic_local_merge<<<NSORT / SEG, 512, 0, stream>>>(keys, k); // j = 512..1
  }

  // ---- Kruskal union-find (acceptance order == descending weight order) ----
  k_kruskal<<<1, 1, 0, stream>>>(keys, parent, rnk, mi, mj);

  // ---- outputs ----
  k_write_edges<<<(NMST + 255) / 256, 256, 0, stream>>>(mi, mj, out_edges);
  k_gather_hp<<<(B_N * 2 * H_DIM) / 256, 256, 0, stream>>>(h2, mi, mj, hp);
  k_gemm_wmma_f32<<<dim3(B_N / 16, H_DIM / 16), 32, 0, stream>>>(hp, Wg1, t1, 2 * H_DIM, H_DIM);
  k_bias_relu<<<(B_N * H_DIM) / 256, 256, 0, stream>>>(t1, bg1, t1, H_DIM, B_N * H_DIM);
  k_gemm_wmma_f32<<<dim3(B_N / 16, D_DIM / 16), 32, 0, stream>>>(t1, Wg2, t2, H_DIM, D_DIM);
  k_bias_tanh_store<<<(NMST * D_DIM + 255) / 256, 256, 0, stream>>>(t2, bg2, out_gamma);
}